// GCN_27376121545092
// MI455X (gfx1250) — compile-verified
//
#include <hip/hip_runtime.h>

// ---------------------------------------------------------------------------
// GCN 2-layer forward for MI455X (gfx1250, wave32).
//   h = relu( (D_in^-1/2 A^T (D_out^-1/2 X)) W1 + b1 )
//   y =       (D_in^-1/2 A^T (D_out^-1/2 h)) W2 + b2
// Scatter phase is HBM/atomic bound; GEMM uses fp32 WMMA (v_wmma_f32_16x16x4_f32).
// ---------------------------------------------------------------------------

#define FEAT 128

typedef __attribute__((ext_vector_type(2))) float v2f;
typedef __attribute__((ext_vector_type(8))) float v8f;

// ---------------- zero fill ----------------
__global__ void gcn_zero_kernel(float* __restrict__ p, int n) {
    int i = blockIdx.x * blockDim.x + threadIdx.x;
    if (i < n) p[i] = 0.0f;
}

// ---------------- degree accumulation ----------------
__global__ void gcn_degree_kernel(const int* __restrict__ src,
                                  const int* __restrict__ dst,
                                  float* __restrict__ degOut,
                                  float* __restrict__ degIn,
                                  int E) {
    int e = blockIdx.x * blockDim.x + threadIdx.x;
    if (e < E) {
        atomicAdd(&degOut[src[e]], 1.0f);
        atomicAdd(&degIn[dst[e]], 1.0f);
    }
}

// ---------------- norms = rsqrt(max(deg,1)) ----------------
__global__ void gcn_norm_kernel(const float* __restrict__ degOut,
                                const float* __restrict__ degIn,
                                float* __restrict__ nrmOut,
                                float* __restrict__ nrmIn,
                                int N) {
    int i = blockIdx.x * blockDim.x + threadIdx.x;
    if (i < N) {
        nrmOut[i] = rsqrtf(fmaxf(degOut[i], 1.0f));
        nrmIn[i]  = rsqrtf(fmaxf(degIn[i], 1.0f));
    }
}

// ---------------- xs = x * norm_out[row]  (float4 per thread) ----------------
__global__ void gcn_scale_kernel(const float* __restrict__ x,
                                 const float* __restrict__ nrm,
                                 float* __restrict__ xs,
                                 int N) {
    int idx = blockIdx.x * blockDim.x + threadIdx.x;   // N*32 threads
    int row = idx >> 5;
    if (row < N) {
        int c = (idx & 31) << 2;
        float s = nrm[row];
        const float4 v = *reinterpret_cast<const float4*>(&x[(size_t)row * FEAT + c]);
        float4 o;
        o.x = v.x * s; o.y = v.y * s; o.z = v.z * s; o.w = v.w * s;
        *reinterpret_cast<float4*>(&xs[(size_t)row * FEAT + c]) = o;
    }
}

// ---------------- edge scatter: agg[dst] += xs[src]  (1 wave / edge) --------
__global__ void gcn_scatter_kernel(const int* __restrict__ src,
                                   const int* __restrict__ dst,
                                   const float* __restrict__ xs,
                                   float* __restrict__ agg,
                                   int E) {
    int gtid = blockIdx.x * blockDim.x + threadIdx.x;
    int e    = gtid >> 5;           // 32 lanes per edge
    int lane = gtid & 31;
    if (e < E) {
        int s = src[e];
        int d = dst[e];
        int c = lane << 2;          // 4 floats per lane -> 128 feats
        const float4 v = *reinterpret_cast<const float4*>(&xs[(size_t)s * FEAT + c]);
        float* p = &agg[(size_t)d * FEAT + c];
        atomicAdd(p + 0, v.x);
        atomicAdd(p + 1, v.y);
        atomicAdd(p + 2, v.z);
        atomicAdd(p + 3, v.w);
    }
}

// ---------------- fused: C = (A * norm_in[row]) @ W + b, optional ReLU ------
// A: [N,128] row-major, W: [128,128] row-major (W[k][n]), C: [N,128].
// Block = 256 threads = 8 waves. Block handles 16 rows; wave w handles
// columns [16w, 16w+16). K=128 consumed in 32 steps of v_wmma_f32_16x16x4_f32.
//
// Fragment layouts (ISA 7.12.2, wave32):
//   A 16x4 fp32 : lanes 0-15 -> M=lane,  VGPR0=K0,VGPR1=K1
//                 lanes16-31 -> M=lane-16,VGPR0=K2,VGPR1=K3
//   B 4x16 fp32 : row striped across lanes; VGPR index = K lsb, lane-half = K msb
//   C 16x16 fp32: VGPR v -> M = v + 8*(lane>=16), N = lane&15
#define LDA 132   // padded LDS stride (132 % 64 = 4 -> conflict-free column reads)

__global__ __launch_bounds__(256)
void gcn_gemm_wmma_kernel(const float* __restrict__ A,
                          const float* __restrict__ nrmIn,
                          const float* __restrict__ W,
                          const float* __restrict__ bias,
                          float* __restrict__ C,
                          int relu) {
    __shared__ float As[16 * LDA];

    const int rowBase = blockIdx.x * 16;
    const int t = threadIdx.x;

    // Stage 16x128 A-tile into LDS, pre-scaled by norm_in (512 float4 loads).
    #pragma unroll
    for (int i = 0; i < 2; ++i) {
        int idx4 = t + i * 256;              // 0..511
        int r    = idx4 >> 5;                // 32 float4 per row
        int c    = (idx4 & 31) << 2;
        float s  = nrmIn[rowBase + r];
        const float4 v = *reinterpret_cast<const float4*>(&A[(size_t)(rowBase + r) * FEAT + c]);
        As[r * LDA + c + 0] = v.x * s;
        As[r * LDA + c + 1] = v.y * s;
        As[r * LDA + c + 2] = v.z * s;
        As[r * LDA + c + 3] = v.w * s;
    }
    __syncthreads();

    const int lane   = t & 31;
    const int wave   = t >> 5;       // 0..7 -> N tile
    const int n0     = wave * 16;
    const int laneLo = lane & 15;
    const int laneHi = lane >> 4;    // 0 or 1 (selects K pair)

    v8f acc = {};
    #pragma unroll
    for (int k0 = 0; k0 < FEAT; k0 += 4) {
        const int ka = k0 + 2 * laneHi;
        v2f a, b;
        a.x = As[laneLo * LDA + ka + 0];
        a.y = As[laneLo * LDA + ka + 1];
        b.x = W[(size_t)(ka + 0) * FEAT + n0 + laneLo];
        b.y = W[(size_t)(ka + 1) * FEAT + n0 + laneLo];
        acc = __builtin_amdgcn_wmma_f32_16x16x4_f32(
            /*neg_a=*/false, a, /*neg_b=*/false, b,
            /*c_mod=*/(short)0, acc, /*reuse_a=*/false, /*reuse_b=*/false);
    }

    const int col = n0 + laneLo;
    const float bc = bias[col];
    #pragma unroll
    for (int v = 0; v < 8; ++v) {
        int r = rowBase + v + 8 * laneHi;
        float o = acc[v] + bc;
        if (relu) o = fmaxf(o, 0.0f);
        C[(size_t)r * FEAT + col] = o;
    }
}

// ---------------------------------------------------------------------------
extern "C" void kernel_launch(void* const* d_in, const int* in_sizes, int n_in,
                              void* d_out, int out_size, void* d_ws, size_t ws_size,
                              hipStream_t stream) {
    const float* x   = (const float*)d_in[0];
    const int*   src = (const int*)  d_in[1];
    const int*   dst = (const int*)  d_in[2];
    const float* W1  = (const float*)d_in[3];
    const float* b1  = (const float*)d_in[4];
    const float* W2  = (const float*)d_in[5];
    const float* b2  = (const float*)d_in[6];
    float* out = (float*)d_out;

    const int N = in_sizes[0] / FEAT;   // 50000
    const int E = in_sizes[1];          // 600000

    // workspace layout (floats)
    float* ws     = (float*)d_ws;
    float* degOut = ws;                         // N
    float* degIn  = ws + N;                     // N
    float* nrmOut = ws + 2 * (size_t)N;         // N
    float* nrmIn  = ws + 3 * (size_t)N;         // N
    float* buf0   = ws + 4 * (size_t)N;                       // N*128
    float* buf1   = ws + 4 * (size_t)N + (size_t)N * FEAT;    // N*128

    const int NF = N * FEAT;
    const int TB = 256;
    dim3 blk(TB);

    // ---- degrees + norms (shared by both layers) ----
    gcn_zero_kernel<<<(2 * N + TB - 1) / TB, blk, 0, stream>>>(degOut, 2 * N);
    gcn_degree_kernel<<<(E + TB - 1) / TB, blk, 0, stream>>>(src, dst, degOut, degIn, E);
    gcn_norm_kernel<<<(N + TB - 1) / TB, blk, 0, stream>>>(degOut, degIn, nrmOut, nrmIn, N);

    // ---- layer 1 ----
    gcn_scale_kernel<<<((size_t)N * 32 + TB - 1) / TB, blk, 0, stream>>>(x, nrmOut, buf0, N);
    gcn_zero_kernel<<<(NF + TB - 1) / TB, blk, 0, stream>>>(buf1, NF);
    gcn_scatter_kernel<<<((size_t)E * 32 + TB - 1) / TB, blk, 0, stream>>>(src, dst, buf0, buf1, E);
    gcn_gemm_wmma_kernel<<<N / 16, blk, 0, stream>>>(buf1, nrmIn, W1, b1, buf0, /*relu=*/1);

    // ---- layer 2 ----
    gcn_scale_kernel<<<((size_t)N * 32 + TB - 1) / TB, blk, 0, stream>>>(buf0, nrmOut, buf1, N);
    gcn_zero_kernel<<<(NF + TB - 1) / TB, blk, 0, stream>>>(buf0, NF);
    gcn_scatter_kernel<<<((size_t)E * 32 + TB - 1) / TB, blk, 0, stream>>>(src, dst, buf1, buf0, E);
    gcn_gemm_wmma_kernel<<<N / 16, blk, 0, stream>>>(buf0, nrmIn, W2, b2, out, /*relu=*/0);
}